// SVF_9028021256648
// MI455X (gfx1250) — compile-verified
//
#include <hip/hip_runtime.h>

// CDNA5 / gfx1250, wave32.
typedef __attribute__((ext_vector_type(2))) float v2f;
typedef __attribute__((ext_vector_type(8))) float v8f;

// Cascaded-SVF frequency response:
//   H[n] = (1+i) * prod_k ( (X[n,:] @ beta[:,k]) / (X[n,:] @ alpha[:,k]) )
// X = [1, 1/x, 1/x^2] complex; beta/alpha real [3,K], K = 64.
//
// Strategy: the [N,3]x[3,64] complex-times-real products are done with
// V_WMMA_F32_16X16X4_F32 (K padded 3->4). One wave handles a 16-row tile and
// all 64 sections (4 column tiles x 4 WMMAs). D-matrix layout: lane = section
// column, VGPR v = row v + 8*(lane>=16). Complex division + running product
// are VALU; the across-section product finishes with a 16-lane shfl_xor
// butterfly inside each half-wave.

__global__ __launch_bounds__(256) void svf_wmma_kernel(
    const float* __restrict__ xr, const float* __restrict__ xi,
    const float* __restrict__ f,  const float* __restrict__ R,
    const float* __restrict__ mLP, const float* __restrict__ mBP,
    const float* __restrict__ mHP,
    float* __restrict__ out, int N)
{
    const int lane       = threadIdx.x & 31;
    const int waveInBlk  = threadIdx.x >> 5;
    const int wavesPerBlk = blockDim.x >> 5;
    const int gwave  = blockIdx.x * wavesPerBlk + waveInBlk;
    const int nwaves = gridDim.x * wavesPerBlk;

    const int  r  = lane & 15;          // row (A) / column (B,D) within tile
    const bool hi = lane >= 16;         // K=2,3 half for A/B; rows 8..15 for D

    // ---- Per-wave coefficient setup: B tiles for beta and alpha ----------
    // B (4x16) lane layout: lanes 0-15 hold {K=0,K=1} for column N=lane,
    // lanes 16-31 hold {K=2, K=3(=0 pad)} for column N=lane-16.
    v2f Bb[4], Ba[4];
#pragma unroll
    for (int t = 0; t < 4; ++t) {
        int k = r + 16 * t;
        float fk = f[k], Rk = R[k], lp = mLP[k], bp = mBP[k], hp = mHP[k];
        float f2 = fk * fk;
        float b0 = f2 * lp + fk * bp + hp;
        float b1 = 2.0f * f2 * lp - 2.0f * hp;
        float b2 = f2 * lp - fk * bp + hp;
        float a0 = f2 + 2.0f * Rk * fk + 1.0f;
        float a1 = 2.0f * f2 - 2.0f;
        float a2 = f2 - 2.0f * Rk * fk + 1.0f;
        v2f blo = {b0, b1}, bhi = {b2, 0.0f};
        v2f alo = {a0, a1}, ahi = {a2, 0.0f};
        Bb[t] = hi ? bhi : blo;
        Ba[t] = hi ? ahi : alo;
    }

    const int ntiles = N >> 4;
    for (int rt = gwave; rt < ntiles; rt += nwaves) {
        const int base = rt << 4;

        // Per-lane x for row (base + r); both half-waves load the same 16
        // values (coalesced 64B).
        float x_re = xr[base + r];
        float x_im = xi[base + r];
        float d    = x_re * x_re + x_im * x_im;
        float rc   = __builtin_amdgcn_rcpf(d);
        rc = rc * (2.0f - d * rc);           // Newton refine
        float ir  = x_re * rc;               // Re(1/x)
        float ii  = -x_im * rc;              // Im(1/x)
        float i2r = ir * ir - ii * ii;       // Re(1/x^2)
        float i2i = 2.0f * ir * ii;          // Im(1/x^2)

        // A (16x4): lanes 0-15 hold {K0=1, K1=Re(1/x)}, lanes 16-31 hold
        // {K2=Re(1/x^2), K3=0} for row M = lane%16. Same for the Im part.
        v2f alo_re = {1.0f, ir},  ahi_re = {i2r, 0.0f};
        v2f alo_im = {0.0f, ii},  ahi_im = {i2i, 0.0f};
        v2f Are = hi ? ahi_re : alo_re;
        v2f Aim = hi ? ahi_im : alo_im;

        // Running complex product per row (VGPR v -> row v + 8*hi).
        float pre[8], pim[8];
#pragma unroll
        for (int v = 0; v < 8; ++v) { pre[v] = 1.0f; pim[v] = 0.0f; }

#pragma unroll
        for (int t = 0; t < 4; ++t) {
            v8f z = {};
            v8f nr = __builtin_amdgcn_wmma_f32_16x16x4_f32(
                false, Are, false, Bb[t], (short)0, z, false, false);
            v8f ni = __builtin_amdgcn_wmma_f32_16x16x4_f32(
                false, Aim, false, Bb[t], (short)0, z, false, false);
            v8f dr = __builtin_amdgcn_wmma_f32_16x16x4_f32(
                false, Are, false, Ba[t], (short)0, z, false, false);
            v8f di = __builtin_amdgcn_wmma_f32_16x16x4_f32(
                false, Aim, false, Ba[t], (short)0, z, false, false);

#pragma unroll
            for (int v = 0; v < 8; ++v) {
                float dd = dr[v] * dr[v] + di[v] * di[v];
                float rd = __builtin_amdgcn_rcpf(dd);
                rd = rd * (2.0f - dd * rd);
                float qr = (nr[v] * dr[v] + ni[v] * di[v]) * rd;
                float qi = (ni[v] * dr[v] - nr[v] * di[v]) * rd;
                float tr = pre[v] * qr - pim[v] * qi;
                pim[v]   = pre[v] * qi + pim[v] * qr;
                pre[v]   = tr;
            }
        }

        // Multiply partial products across the 16 lanes of each half-wave
        // (xor masks 1,2,4,8 stay inside each 16-lane group).
#pragma unroll
        for (int m = 1; m <= 8; m <<= 1) {
#pragma unroll
            for (int v = 0; v < 8; ++v) {
                float ore = __shfl_xor(pre[v], m, 32);
                float oim = __shfl_xor(pim[v], m, 32);
                float tr  = pre[v] * ore - pim[v] * oim;
                pim[v]    = pre[v] * oim + pim[v] * ore;
                pre[v]    = tr;
            }
        }

        // Extract p[r] without dynamic register indexing (cndmask chain),
        // then lanes r<8 of each half write rows base+8*hi+r.
        float sre = pre[0], sim = pim[0];
#pragma unroll
        for (int v = 1; v < 8; ++v) {
            if (r == v) { sre = pre[v]; sim = pim[v]; }
        }
        if (r < 8) {
            int row = base + (hi ? 8 : 0) + r;
            // H = (1 + i) * P
            float2 o;
            o.x = sre - sim;
            o.y = sre + sim;
            reinterpret_cast<float2*>(out)[row] = o;
        }
    }
}

extern "C" void kernel_launch(void* const* d_in, const int* in_sizes, int n_in,
                              void* d_out, int out_size, void* d_ws, size_t ws_size,
                              hipStream_t stream) {
    const float* xr  = (const float*)d_in[0];
    const float* xi  = (const float*)d_in[1];
    const float* f   = (const float*)d_in[2];
    const float* R   = (const float*)d_in[3];
    const float* mLP = (const float*)d_in[4];
    const float* mBP = (const float*)d_in[5];
    const float* mHP = (const float*)d_in[6];
    float* out = (float*)d_out;

    const int N = in_sizes[0];          // 1048576 (multiple of 16)
    const int ntiles = N >> 4;
    const int threads = 256;            // 8 waves per block
    const int wavesPerBlk = threads / 32;
    // ~8 row-tiles per wave to amortize the coefficient setup.
    int blocks = (ntiles + wavesPerBlk * 8 - 1) / (wavesPerBlk * 8);
    if (blocks > 1024) blocks = 1024;
    if (blocks < 1)    blocks = 1;

    svf_wmma_kernel<<<blocks, threads, 0, stream>>>(
        xr, xi, f, R, mLP, mBP, mHP, out, N);
}